// YOLO_38723425140753
// MI455X (gfx1250) — compile-verified
//
#include <hip/hip_runtime.h>
#include <hip/hip_bf16.h>
#include <cstdint>

#define TOPK     128
#define NCLS     80
#define NB       4096
#define CBUF     512
#define NTHREADS 256
#define NPER     7581          // cells per anchor (5776+1444+361)

typedef __attribute__((ext_vector_type(2))) float v2f;
typedef __attribute__((ext_vector_type(8))) float v8f;

__constant__ float c_anc[9][2] = {
    {12.f, 16.f}, {19.f, 36.f}, {40.f, 28.f},      // level 0 (stride 8)
    {36.f, 75.f}, {76.f, 55.f}, {72.f, 146.f},     // level 1 (stride 16)
    {142.f, 110.f}, {192.f, 243.f}, {459.f, 401.f} // level 2 (stride 32)
};

__device__ __forceinline__ float sigmoidf_(float x) {
    return 1.0f / (1.0f + __expf(-x));
}

__global__ __launch_bounds__(NTHREADS)
void yolo_nms_kernel(const float* __restrict__ p0,
                     const float* __restrict__ p1,
                     const float* __restrict__ p2,
                     float* __restrict__ out)
{
    const int tid = threadIdx.x;
    const int b   = blockIdx.x / NCLS;
    const int cls = blockIdx.x % NCLS;

    __shared__ int      s_hist[NB];          // 16 KB
    __shared__ int      s_chunk[NTHREADS];
    __shared__ float    s_cval[CBUF];
    __shared__ int      s_cidx[CBUF];
    __shared__ int      s_cnt;
    __shared__ int      s_bstar;
    __shared__ float    s_bx[TOPK][4];
    __shared__ float    s_area[TOPK];
    __shared__ unsigned s_supp[TOPK][4];     // 128x128 suppression bits
    __shared__ unsigned s_keep[4];
    unsigned* s_suppF = &s_supp[0][0];

    for (int i = tid; i < NB; i += NTHREADS) s_hist[i] = 0;
    if (tid == 0) s_cnt = 0;
    __syncthreads();

    const float binScale = (float)NB / 0.7f;

    // Visit every candidate's prob for this (b, cls); calls fn(prob, n) for prob > 0.3
    auto forEach = [&](auto&& fn) {
        #pragma unroll
        for (int l = 0; l < 3; ++l) {
            const int hw  = (l == 0) ? 5776 : ((l == 1) ? 1444 : 361);
            const int off = (l == 0) ? 0    : ((l == 1) ? 5776 : 7220);
            const float* pl = (l == 0) ? p0 : ((l == 1) ? p1 : p2);
            const float* base = pl + (size_t)b * 255 * hw;
            #pragma unroll
            for (int a = 0; a < 3; ++a) {
                const float* po = base + (size_t)(a * 85 + 4) * hw;
                const float* pc = base + (size_t)(a * 85 + 5 + cls) * hw;
                const int nb = a * NPER + off;
                for (int cell = tid; cell < hw; cell += NTHREADS) {
                    float pr = sigmoidf_(po[cell]) * sigmoidf_(pc[cell]);
                    if (pr > 0.3f) fn(pr, nb + cell);
                }
            }
        }
    };

    // ---- Pass 1: histogram of probs in (0.3, 1] ----
    forEach([&](float pr, int) {
        int bb = (int)((pr - 0.3f) * binScale);
        bb = bb < 0 ? 0 : (bb > NB - 1 ? NB - 1 : bb);
        atomicAdd(&s_hist[bb], 1);
    });
    __syncthreads();

    // chunk sums (16 bins per thread) + pad candidate buffer
    {
        int s = 0;
        #pragma unroll
        for (int q = 0; q < 16; ++q) s += s_hist[tid * 16 + q];
        s_chunk[tid] = s;
        s_cval[tid]            = -1.0f; s_cidx[tid]            = 0x7FFFFFFF;
        s_cval[tid + NTHREADS] = -1.0f; s_cidx[tid + NTHREADS] = 0x7FFFFFFF;
    }
    __syncthreads();

    if (tid == 0) {
        int cum = 0, bstar = 0; bool found = false;
        for (int cch = NTHREADS - 1; cch >= 0 && !found; --cch) {
            if (cum + s_chunk[cch] >= TOPK) {
                for (int bb = cch * 16 + 15; bb >= cch * 16; --bb) {
                    cum += s_hist[bb];
                    if (cum >= TOPK) { bstar = bb; found = true; break; }
                }
            } else cum += s_chunk[cch];
        }
        s_bstar = found ? bstar : 0;
    }
    __syncthreads();

    // ---- Pass 2: collect candidates at/above the cutoff bin ----
    const int bstar = s_bstar;
    forEach([&](float pr, int n) {
        int bb = (int)((pr - 0.3f) * binScale);
        bb = bb < 0 ? 0 : (bb > NB - 1 ? NB - 1 : bb);
        if (bb >= bstar) {
            int pos = atomicAdd(&s_cnt, 1);
            if (pos < CBUF) { s_cval[pos] = pr; s_cidx[pos] = n; }
        }
    });
    __syncthreads();

    // ---- Bitonic sort CBUF slots: score desc, index asc on ties ----
    for (int kk = 2; kk <= CBUF; kk <<= 1) {
        for (int jj = kk >> 1; jj > 0; jj >>= 1) {
            for (int i = tid; i < CBUF; i += NTHREADS) {
                int ixj = i ^ jj;
                if (ixj > i) {
                    float v1 = s_cval[i], v2 = s_cval[ixj];
                    int   i1 = s_cidx[i], i2 = s_cidx[ixj];
                    bool betterIxj = (v2 > v1) || (v2 == v1 && i2 < i1);
                    bool up = ((i & kk) == 0);
                    if (betterIxj == up) {
                        s_cval[i] = v2; s_cval[ixj] = v1;
                        s_cidx[i] = i2; s_cidx[ixj] = i1;
                    }
                }
            }
            __syncthreads();
        }
    }

    // ---- Decode boxes for the 128 selected; zero suppression bits ----
    for (int i = tid; i < TOPK * 4; i += NTHREADS) s_suppF[i] = 0u;
    if (tid < TOPK) {
        int n = s_cidx[tid];
        if ((unsigned)n >= (unsigned)(3 * NPER)) n = 0;  // padding slot
        int a = n / NPER;
        int m = n - a * NPER;
        int l, w, off;
        if (m < 5776)      { l = 0; w = 76; off = 0;    }
        else if (m < 7220) { l = 1; w = 38; off = 5776; }
        else               { l = 2; w = 19; off = 7220; }
        int cell = m - off;
        int hw = w * w;
        const float* pl = (l == 0) ? p0 : ((l == 1) ? p1 : p2);
        const float* base = pl + ((size_t)b * 255 + a * 85) * hw;
        float tx = base[cell];
        float ty = base[hw + cell];
        float tw = base[2 * hw + cell];
        float th = base[3 * hw + cell];
        int gx = cell % w, gy = cell / w;
        float invw = 1.0f / (float)w;
        float cx = (sigmoidf_(tx) + (float)gx) * invw;
        float cy = (sigmoidf_(ty) + (float)gy) * invw;
        // anchor/(red*grid) == anchor/608 for every level (8*76=16*38=32*19=608)
        float bw = __expf(tw) * c_anc[l * 3 + a][0] * (1.0f / 608.0f);
        float bh = __expf(th) * c_anc[l * 3 + a][1] * (1.0f / 608.0f);
        float x1 = cx - 0.5f * bw, y1 = cy - 0.5f * bh;
        s_bx[tid][0] = x1; s_bx[tid][1] = y1;
        s_bx[tid][2] = x1 + bw; s_bx[tid][3] = y1 + bh;
        s_area[tid] = bw * bh;
    }
    __syncthreads();

    // ---- IoU via WMMA (pairwise area-sum = rank-2 matmul) + VALU intersection ----
    {
        const int lane = tid & 31;
        const int wv   = tid >> 5;          // 8 waves
        const int half = lane >> 4;         // K-half for 16x16x4 f32 layout
        const int l16  = lane & 15;
        const float hm = (half == 0) ? 1.0f : 0.0f;   // branch-free operand mask
        for (int t = wv; t < 64; t += 8) {  // wave-uniform: EXEC all-ones at WMMA
            const int ti = t >> 3, tj = t & 7;
            // unconditional LDS loads; zero the K=2/3 half by multiply, not branch
            const float ai = s_area[ti * 16 + l16];
            const float aj = s_area[tj * 16 + l16];
            v2f av, bv;
            // A (16x4): col0 = area_i, col1 = 1, cols 2..3 = 0 (lanes 16..31)
            av.x = ai * hm;  av.y = hm;
            // B (4x16): row0 = 1, row1 = area_j, rows 2..3 = 0
            bv.x = hm;       bv.y = aj * hm;
            v8f cz = {};
            // D[i][j] = area_i + area_j
            v8f d = __builtin_amdgcn_wmma_f32_16x16x4_f32(
                false, av, false, bv, (short)0, cz, false, false);
            const int gj = tj * 16 + l16;
            const float xj1 = s_bx[gj][0], yj1 = s_bx[gj][1];
            const float xj2 = s_bx[gj][2], yj2 = s_bx[gj][3];
            #pragma unroll
            for (int r = 0; r < 8; ++r) {
                const int gi = ti * 16 + r + half * 8;
                float iw = fminf(s_bx[gi][2], xj2) - fmaxf(s_bx[gi][0], xj1);
                float ih = fminf(s_bx[gi][3], yj2) - fmaxf(s_bx[gi][1], yj1);
                iw = fmaxf(iw, 0.0f); ih = fmaxf(ih, 0.0f);
                float inter = iw * ih;
                // iou > 0.45  <=>  inter > 0.45*(union); union >= 0, NaN/0 cases agree
                if (inter > 0.45f * (d[r] - inter))
                    atomicOr(&s_suppF[gi * 4 + (gj >> 5)], 1u << (gj & 31));
            }
        }
    }
    __syncthreads();

    // ---- Initial keep mask (score > 0) ----
    if (tid < 4) {
        unsigned m = 0;
        #pragma unroll
        for (int q = 0; q < 32; ++q)
            if (s_cval[tid * 32 + q] > 0.0f) m |= 1u << q;
        s_keep[tid] = m;
    }
    __syncthreads();

    // ---- Greedy sequential suppression (matches lax.scan step) ----
    if (tid == 0) {
        for (int i = 0; i < TOPK; ++i) {
            const int wi = i >> 5, bi = i & 31;
            if ((s_keep[wi] >> bi) & 1u) {
                #pragma unroll
                for (int w = 0; w < 4; ++w) {
                    unsigned m = s_supp[i][w];
                    if (w < wi) m = 0u;
                    else if (w == wi) m &= ~((2u << bi) - 1u);  // only j > i
                    s_keep[w] &= ~m;
                }
            }
        }
    }
    __syncthreads();

    // ---- Write output [b, C, K, 6] ----
    if (tid < TOPK) {
        const int k = tid;
        const bool kept = (s_keep[k >> 5] >> (k & 31)) & 1u;
        float* o = out + ((size_t)(b * NCLS + cls) * TOPK + k) * 6;
        float x1 = 0.f, y1 = 0.f, x2 = 0.f, y2 = 0.f, sc = 0.f;
        if (kept) {
            x1 = s_bx[k][0]; y1 = s_bx[k][1];
            x2 = s_bx[k][2]; y2 = s_bx[k][3];
            sc = s_cval[k];
        }
        o[0] = x1; o[1] = y1; o[2] = x2; o[3] = y2;
        o[4] = sc; o[5] = (float)cls;    // class id written even for empty rows
    }
}

extern "C" void kernel_launch(void* const* d_in, const int* in_sizes, int n_in,
                              void* d_out, int out_size, void* d_ws, size_t ws_size,
                              hipStream_t stream) {
    (void)n_in; (void)d_ws; (void)ws_size; (void)out_size;
    const float* p0 = (const float*)d_in[0];  // [b,255,76,76]
    const float* p1 = (const float*)d_in[1];  // [b,255,38,38]
    const float* p2 = (const float*)d_in[2];  // [b,255,19,19]
    float* out = (float*)d_out;               // [b,80,128,6]
    const int batch = in_sizes[0] / (255 * 76 * 76);
    yolo_nms_kernel<<<dim3(batch * NCLS), dim3(NTHREADS), 0, stream>>>(p0, p1, p2, out);
}